// GSF_DTA_46308337385757
// MI455X (gfx1250) — compile-verified
//
#include <hip/hip_runtime.h>

typedef __attribute__((ext_vector_type(2))) float v2f;
typedef __attribute__((ext_vector_type(8))) float v8f;

// ---------------------------------------------------------------------------
// Utility kernels
// ---------------------------------------------------------------------------
__global__ void fill_f_k(float* __restrict__ p, float v, long long n) {
  long long i = (long long)blockIdx.x * blockDim.x + threadIdx.x;
  if (i < n) p[i] = v;
}

__global__ void fill_i_k(int* __restrict__ p, int v, long long n) {
  long long i = (long long)blockIdx.x * blockDim.x + threadIdx.x;
  if (i < n) p[i] = v;
}

__global__ void copy_i_k(const int* __restrict__ src, int* __restrict__ dst, long long n) {
  long long i = (long long)blockIdx.x * blockDim.x + threadIdx.x;
  if (i < n) dst[i] = src[i];
}

// ---------------------------------------------------------------------------
// CSR build: histogram of in-degrees -> exclusive scan -> cursor scatter
// ---------------------------------------------------------------------------
__global__ void hist_k(const int* __restrict__ dst, int* __restrict__ count, long long E) {
  long long e = (long long)blockIdx.x * blockDim.x + threadIdx.x;
  if (e < E) atomicAdd(&count[dst[e]], 1);
}

// dinv = rsqrt(1 + in_count)  (self-loop adds 1, degree always > 0)
__global__ void dinv_k(const int* __restrict__ count, float* __restrict__ dinv, long long n) {
  long long i = (long long)blockIdx.x * blockDim.x + threadIdx.x;
  if (i < n) dinv[i] = rsqrtf(1.0f + (float)count[i]);
}

// Block-level exclusive scan (Hillis-Steele in LDS), block totals to bsum.
__global__ void scan_block_k(const int* __restrict__ in, int* __restrict__ out,
                             int* __restrict__ bsum, int n) {
  __shared__ int tmp[256];
  const int tid = threadIdx.x;
  const int g = blockIdx.x * 256 + tid;
  const int v = (g < n) ? in[g] : 0;
  tmp[tid] = v;
  __syncthreads();
  for (int off = 1; off < 256; off <<= 1) {
    int t = (tid >= off) ? tmp[tid - off] : 0;
    __syncthreads();
    tmp[tid] += t;
    __syncthreads();
  }
  if (g < n) out[g] = tmp[tid] - v;  // exclusive
  if (tid == 255) bsum[blockIdx.x] = tmp[255];
}

// Serial exclusive scan of block sums (nb ~ 391, trivial).
__global__ void scan_sums_k(int* __restrict__ bsum, int nb) {
  if (blockIdx.x == 0 && threadIdx.x == 0) {
    int acc = 0;
    for (int i = 0; i < nb; ++i) { int t = bsum[i]; bsum[i] = acc; acc += t; }
  }
}

__global__ void add_off_k(int* __restrict__ out, const int* __restrict__ bsum, int n) {
  const int g = blockIdx.x * 256 + threadIdx.x;
  if (g < n) out[g] += bsum[blockIdx.x];
}

// Scatter edge ids into CSR slots (integer atomics only).
__global__ void scatter_k(const int* __restrict__ dst, int* __restrict__ cursor,
                          int* __restrict__ eidx, long long E) {
  long long e = (long long)blockIdx.x * blockDim.x + threadIdx.x;
  if (e < E) {
    const int pos = atomicAdd(&cursor[dst[e]], 1);
    eidx[pos] = (int)e;
  }
}

// ---------------------------------------------------------------------------
// Y[N x 128] = X[N x K] @ W[K x 128], f32 WMMA 16x16x4.
// One block = 256 threads = 8 waves; block covers 16 rows x 128 cols.
// A tile (16 x K) staged in LDS (padded), shared by all 8 waves.
// ---------------------------------------------------------------------------
template <int K>
__global__ void gemm_xw_wmma(const float* __restrict__ X, const float* __restrict__ W,
                             float* __restrict__ Y, long long N) {
  __shared__ float ldsA[16 * (K + 1)];
  const int tid  = threadIdx.x;        // 0..255
  const int wave = tid >> 5;           // 0..7  -> column tile
  const int lane = tid & 31;
  const long long row0 = (long long)blockIdx.x * 16;

  for (int i = tid; i < 16 * K; i += 256) {
    const int r = i / K, c = i % K;
    ldsA[r * (K + 1) + c] = X[(row0 + r) * K + c];
  }
  __syncthreads();

  const int half = lane >> 4;          // 0: K+0/K+1,  1: K+2/K+3
  const int l16  = lane & 15;
  const int col  = wave * 16 + l16;    // 0..127

  v8f acc = {};
  for (int k = 0; k < K; k += 4) {
    const int kk = k + half * 2;
    v2f a, b;
    a.x = ldsA[l16 * (K + 1) + kk];
    a.y = ldsA[l16 * (K + 1) + kk + 1];
    b.x = W[(long long)kk * 128 + col];
    b.y = W[(long long)(kk + 1) * 128 + col];
    acc = __builtin_amdgcn_wmma_f32_16x16x4_f32(
        false, a, false, b, (short)0, acc, false, false);
  }

#pragma unroll
  for (int r = 0; r < 8; ++r) {
    const long long m = row0 + r + half * 8;
    Y[m * 128 + col] = acc[r];
  }
}

// ---------------------------------------------------------------------------
// CSR gather aggregation, fused with self-loop + bias + ReLU.
// One wave per destination node; 32 lanes x float4 = 128 features in VGPRs.
// H[d] = relu( sum_e XW[src(e)]*dinv[s]*dinv[d] + XW[d]*dinv[d]^2 + bias )
// Zero scatter atomics; output written exactly once.
// ---------------------------------------------------------------------------
__global__ void gcn_gather_k(const float* __restrict__ XW, const int* __restrict__ src,
                             const int* __restrict__ rowptr, const int* __restrict__ count,
                             const int* __restrict__ eidx, const float* __restrict__ dinv,
                             const float* __restrict__ bias, float* __restrict__ H,
                             long long N) {
  const int wave = threadIdx.x >> 5;   // 0..7
  const int lane = threadIdx.x & 31;
  const long long d = (long long)blockIdx.x * 8 + wave;
  if (d >= N) return;

  const float dd  = dinv[d];
  const int start = rowptr[d];
  const int cnt   = count[d];

  float4 acc = make_float4(0.f, 0.f, 0.f, 0.f);
  for (int i = 0; i < cnt; ++i) {
    const int e = eidx[start + i];
    const long long s = (long long)src[e];
    const float nrm = dinv[s] * dd;
    const float4 v = *(const float4*)(XW + s * 128 + lane * 4);
    acc.x += v.x * nrm;
    acc.y += v.y * nrm;
    acc.z += v.z * nrm;
    acc.w += v.w * nrm;
  }

  const float4 xv = *(const float4*)(XW + d * 128 + lane * 4);
  const float sl = dd * dd;
  const int cb = lane * 4;
  float4 o;
  o.x = acc.x + xv.x * sl + bias[cb + 0];
  o.y = acc.y + xv.y * sl + bias[cb + 1];
  o.z = acc.z + xv.z * sl + bias[cb + 2];
  o.w = acc.w + xv.w * sl + bias[cb + 3];
  o.x = o.x > 0.f ? o.x : 0.f;
  o.y = o.y > 0.f ? o.y : 0.f;
  o.z = o.z > 0.f ? o.z : 0.f;
  o.w = o.w > 0.f ? o.w : 0.f;
  *(float4*)(H + d * 128 + lane * 4) = o;
}

// pooled[col] += mean over rows (atomic partials; pooled pre-zeroed)
__global__ void mean_pool_k(const float* __restrict__ H, float* __restrict__ pooled,
                            long long N, float invN) {
  __shared__ float red[256];
  const int tid = threadIdx.x;
  const int col = tid & 127;
  const long long rbase = (long long)blockIdx.x * 512;
  long long rend = rbase + 512;
  if (rend > N) rend = N;
  float s = 0.f;
  for (long long r = rbase + (tid >> 7); r < rend; r += 2) s += H[r * 128 + col];
  red[tid] = s;
  __syncthreads();
  if (tid < 128) atomicAdd(&pooled[col], (red[tid] + red[tid + 128]) * invN);
}

// out[128] = relu(relu(x @ w1 + b1) @ w2 + b2). One block, 128 threads.
__global__ void seq_encode_k(const float* __restrict__ x,
                             const float* __restrict__ w1, const float* __restrict__ b1,
                             const float* __restrict__ w2, const float* __restrict__ b2,
                             float* __restrict__ out, int K1) {
  __shared__ float h[128];
  const int j = threadIdx.x;
  float s = 0.f;
  for (int k = 0; k < K1; ++k) s += x[k] * w1[k * 128 + j];
  s += b1[j];
  h[j] = s > 0.f ? s : 0.f;
  __syncthreads();
  float t = 0.f;
  for (int k = 0; k < 128; ++k) t += h[k] * w2[k * 128 + j];
  t += b2[j];
  out[j] = t > 0.f ? t : 0.f;
}

// out[0] = relu(comb @ fc1 + b1) @ fc2 + b2, comb = pooled[512].
__global__ void final_mlp_k(const float* __restrict__ pooled,
                            const float* __restrict__ fc1w, const float* __restrict__ fc1b,
                            const float* __restrict__ fc2w, const float* __restrict__ fc2b,
                            float* __restrict__ out) {
  __shared__ float comb[512];
  __shared__ float red[128];
  const int j = threadIdx.x;
  for (int i = j; i < 512; i += 128) comb[i] = pooled[i];
  __syncthreads();
  float s = 0.f;
  for (int k = 0; k < 512; ++k) s += comb[k] * fc1w[k * 128 + j];
  s += fc1b[j];
  s = s > 0.f ? s : 0.f;
  red[j] = s * fc2w[j];
  __syncthreads();
  for (int st = 64; st > 0; st >>= 1) {
    if (j < st) red[j] += red[j + st];
    __syncthreads();
  }
  if (j == 0) out[0] = red[0] + fc2b[0];
}

// ---------------------------------------------------------------------------
// Host orchestration
// ---------------------------------------------------------------------------
static inline unsigned cdiv_u(long long a, long long b) { return (unsigned)((a + b - 1) / b); }

extern "C" void kernel_launch(void* const* d_in, const int* in_sizes, int n_in,
                              void* d_out, int out_size, void* d_ws, size_t ws_size,
                              hipStream_t stream) {
  (void)n_in; (void)out_size; (void)ws_size;

  const float* protein_x = (const float*)d_in[0];
  const int*   p_ei      = (const int*)d_in[1];
  const float* drug_x    = (const float*)d_in[2];
  const int*   d_ei      = (const int*)d_in[3];
  const float* pseq_x    = (const float*)d_in[4];
  const float* dseq_x    = (const float*)d_in[5];
  const float* pc1_w = (const float*)d_in[6],  *pc1_b = (const float*)d_in[7];
  const float* pc2_w = (const float*)d_in[8],  *pc2_b = (const float*)d_in[9];
  const float* dc1_w = (const float*)d_in[10], *dc1_b = (const float*)d_in[11];
  const float* dc2_w = (const float*)d_in[12], *dc2_b = (const float*)d_in[13];
  const float* ps_w1 = (const float*)d_in[14], *ps_b1 = (const float*)d_in[15];
  const float* ps_w2 = (const float*)d_in[16], *ps_b2 = (const float*)d_in[17];
  const float* ds_w1 = (const float*)d_in[18], *ds_b1 = (const float*)d_in[19];
  const float* ds_w2 = (const float*)d_in[20], *ds_b2 = (const float*)d_in[21];
  const float* fc1_w = (const float*)d_in[22], *fc1_b = (const float*)d_in[23];
  const float* fc2_w = (const float*)d_in[24], *fc2_b = (const float*)d_in[25];

  const long long Np = in_sizes[0] / 128;
  const long long Ep = in_sizes[1] / 2;
  const long long Nd = in_sizes[2] / 64;
  const long long Ed = in_sizes[3] / 2;
  const int Psd = in_sizes[4];
  const int Dsd = in_sizes[5];

  const int* p_src = p_ei;
  const int* p_dst = p_ei + Ep;
  const int* d_src = d_ei;
  const int* d_dst = d_ei + Ed;

  // ---- workspace layout ----
  float* wsf = (float*)d_ws;
  float* p_dinv = wsf;  wsf += Np;
  float* d_dinv = wsf;  wsf += Nd;
  float* pA     = wsf;  wsf += Np * 128;   // xw scratch
  float* pB     = wsf;  wsf += Np * 128;   // hidden
  float* dA     = wsf;  wsf += Nd * 128;
  float* dB     = wsf;  wsf += Nd * 128;
  float* pool   = wsf;  wsf += 512;        // [p_pool, d_pool, pseq, dseq]
  int* wsi = (int*)wsf;
  int* p_count  = wsi;  wsi += Np;
  int* p_rowptr = wsi;  wsi += Np;
  int* p_cursor = wsi;  wsi += Np;
  int* p_eidx   = wsi;  wsi += Ep;
  int* d_count  = wsi;  wsi += Nd;
  int* d_rowptr = wsi;  wsi += Nd;
  int* d_cursor = wsi;  wsi += Nd;
  int* d_eidx   = wsi;  wsi += Ed;
  int* bsum     = wsi;  // scan block sums (reused per graph, sequential use)

  const dim3 B256(256);
  const unsigned p_nb = cdiv_u(Np, 256);
  const unsigned d_nb = cdiv_u(Nd, 256);

  // ---- CSR build + norms: protein ----
  fill_i_k<<<p_nb, B256, 0, stream>>>(p_count, 0, Np);
  hist_k<<<cdiv_u(Ep, 256), B256, 0, stream>>>(p_dst, p_count, Ep);
  dinv_k<<<p_nb, B256, 0, stream>>>(p_count, p_dinv, Np);
  scan_block_k<<<p_nb, B256, 0, stream>>>(p_count, p_rowptr, bsum, (int)Np);
  scan_sums_k<<<1, dim3(32), 0, stream>>>(bsum, (int)p_nb);
  add_off_k<<<p_nb, B256, 0, stream>>>(p_rowptr, bsum, (int)Np);
  copy_i_k<<<p_nb, B256, 0, stream>>>(p_rowptr, p_cursor, Np);
  scatter_k<<<cdiv_u(Ep, 256), B256, 0, stream>>>(p_dst, p_cursor, p_eidx, Ep);

  // ---- CSR build + norms: drug ----
  fill_i_k<<<d_nb, B256, 0, stream>>>(d_count, 0, Nd);
  hist_k<<<cdiv_u(Ed, 256), B256, 0, stream>>>(d_dst, d_count, Ed);
  dinv_k<<<d_nb, B256, 0, stream>>>(d_count, d_dinv, Nd);
  scan_block_k<<<d_nb, B256, 0, stream>>>(d_count, d_rowptr, bsum, (int)Nd);
  scan_sums_k<<<1, dim3(32), 0, stream>>>(bsum, (int)d_nb);
  add_off_k<<<d_nb, B256, 0, stream>>>(d_rowptr, bsum, (int)Nd);
  copy_i_k<<<d_nb, B256, 0, stream>>>(d_rowptr, d_cursor, Nd);
  scatter_k<<<cdiv_u(Ed, 256), B256, 0, stream>>>(d_dst, d_cursor, d_eidx, Ed);

  // ---- protein GCN layers (GEMM via WMMA, CSR gather w/ fused epilogue) ----
  gemm_xw_wmma<128><<<cdiv_u(Np, 16), B256, 0, stream>>>(protein_x, pc1_w, pA, Np);
  gcn_gather_k<<<cdiv_u(Np, 8), B256, 0, stream>>>(pA, p_src, p_rowptr, p_count, p_eidx,
                                                   p_dinv, pc1_b, pB, Np);
  gemm_xw_wmma<128><<<cdiv_u(Np, 16), B256, 0, stream>>>(pB, pc2_w, pA, Np);
  gcn_gather_k<<<cdiv_u(Np, 8), B256, 0, stream>>>(pA, p_src, p_rowptr, p_count, p_eidx,
                                                   p_dinv, pc2_b, pB, Np);

  // ---- drug GCN layers ----
  gemm_xw_wmma<64><<<cdiv_u(Nd, 16), B256, 0, stream>>>(drug_x, dc1_w, dA, Nd);
  gcn_gather_k<<<cdiv_u(Nd, 8), B256, 0, stream>>>(dA, d_src, d_rowptr, d_count, d_eidx,
                                                   d_dinv, dc1_b, dB, Nd);
  gemm_xw_wmma<128><<<cdiv_u(Nd, 16), B256, 0, stream>>>(dB, dc2_w, dA, Nd);
  gcn_gather_k<<<cdiv_u(Nd, 8), B256, 0, stream>>>(dA, d_src, d_rowptr, d_count, d_eidx,
                                                   d_dinv, dc2_b, dB, Nd);

  // ---- mean pool ----
  fill_f_k<<<1, B256, 0, stream>>>(pool, 0.0f, 256);
  mean_pool_k<<<cdiv_u(Np, 512), B256, 0, stream>>>(pB, pool + 0,   Np, 1.0f / (float)Np);
  mean_pool_k<<<cdiv_u(Nd, 512), B256, 0, stream>>>(dB, pool + 128, Nd, 1.0f / (float)Nd);

  // ---- sequence encoders ----
  seq_encode_k<<<1, dim3(128), 0, stream>>>(pseq_x, ps_w1, ps_b1, ps_w2, ps_b2, pool + 256, Psd);
  seq_encode_k<<<1, dim3(128), 0, stream>>>(dseq_x, ds_w1, ds_b1, ds_w2, ds_b2, pool + 384, Dsd);

  // ---- final MLP ----
  final_mlp_k<<<1, dim3(128), 0, stream>>>(pool, fc1_w, fc1_b, fc2_w, fc2_b, (float*)d_out);
}